// PressureGNN_18348100288783
// MI455X (gfx1250) — compile-verified
//
#include <hip/hip_runtime.h>
#include <hip/hip_bf16.h>

// ---------------------------------------------------------------------------
// 2-layer GCN forward for MI455X (gfx1250, wave32).
//   h   = relu( Agg(norm, X@W1) + b1 )
//   out = mask ? 0 : Agg(norm, h@W2) + b2
// Agg(norm, H)[d] = sum_{e:(s->d)} dinv[s]*dinv[d]*H[s]  + dinv[d]^2*H[d]
//
// Fusions vs naive pipeline:
//  - GEMM1 epilogue writes both H and the self-loop term agg = dinv^2 * H.
//  - bias+ReLU folded into the layer-2 per-node dot (h_relu never materialized).
// ---------------------------------------------------------------------------

typedef __attribute__((ext_vector_type(2))) float v2f;
typedef __attribute__((ext_vector_type(8))) float v8f;

#define IN_DIM 128
#define HID_DIM 128

// ---------------- degree / normalization ----------------------------------
__global__ void k_deg_init(float* __restrict__ deg, int n) {
  int i = blockIdx.x * blockDim.x + threadIdx.x;
  if (i < n) deg[i] = 1.0f;  // self-loop contribution
}

__global__ void k_deg_accum(const int* __restrict__ dst, float* __restrict__ deg, int E) {
  int e = blockIdx.x * blockDim.x + threadIdx.x;
  if (e < E) atomicAdd(deg + dst[e], 1.0f);
}

__global__ void k_rsqrt_inplace(float* __restrict__ d, int n) {
  int i = blockIdx.x * blockDim.x + threadIdx.x;
  if (i < n) d[i] = rsqrtf(d[i]);  // deg >= 1 always (self-loops)
}

// ---------------- GEMM1: H = X @ W1 via V_WMMA_F32_16X16X4_F32 -------------
// One wave handles a 16-row block of X and all 8 column tiles of W1.
// A (16x4 f32): lanes 0-15 hold M=0..15 {K=0 in v0, K=1 in v1};
//               lanes 16-31 hold M=0..15 {K=2, K=3}.
// B (4x16 f32): lanes 0-15 hold N=0..15 {K=0 in v0, K=1 in v1};
//               lanes 16-31 hold N=0..15 {K=2, K=3}.
// C/D (16x16 f32, 8 VGPRs): VGPR r = row r (lanes 0-15) / row r+8 (lanes 16-31).
// Epilogue stores H and agg = dinv[row]^2 * H (self-loop term of layer-1 agg).
__global__ void __launch_bounds__(128)
k_gemm_x_w1(const float* __restrict__ X, const float* __restrict__ W,
            const float* __restrict__ dinv, float* __restrict__ H,
            float* __restrict__ agg, int N) {
  const int lane = threadIdx.x & 31;
  const int wave = threadIdx.x >> 5;
  const int rowTile = blockIdx.x * 4 + wave;  // 16-row block index (wave-uniform)
  const int m0 = rowTile * 16;
  if (m0 >= N) return;  // wave-uniform exit: EXEC stays all-ones for WMMA
                        // (N is a multiple of 16 here: 100000 = 6250*16)

  const int half = lane >> 4;  // 0: K pair {0,1}, 1: K pair {2,3}
  const int l15 = lane & 15;

  v8f c[8];
#pragma unroll
  for (int t = 0; t < 8; ++t) c[t] = (v8f){0.f, 0.f, 0.f, 0.f, 0.f, 0.f, 0.f, 0.f};

  const float* arow = X + (size_t)(m0 + l15) * IN_DIM;

#pragma unroll 4
  for (int kb = 0; kb < IN_DIM; kb += 4) {
    // A fragment: float2 {K, K+1} starting at kb + 2*half
    const float2 af = *(const float2*)(arow + kb + 2 * half);
    v2f a;
    a.x = af.x;
    a.y = af.y;
#pragma unroll
    for (int t = 0; t < 8; ++t) {
      const int col = t * 16 + l15;
      v2f b;
      b.x = W[(size_t)(kb + 2 * half) * HID_DIM + col];
      b.y = W[(size_t)(kb + 2 * half + 1) * HID_DIM + col];
      // 8 args: (neg_a, A, neg_b, B, c_mod, C, reuse_a, reuse_b)
      c[t] = __builtin_amdgcn_wmma_f32_16x16x4_f32(
          false, a, false, b, (short)0, c[t], false, false);
    }
  }

  // per-lane self-loop scales for the 8 rows this lane's D fragment covers
  float w2s[8];
#pragma unroll
  for (int r = 0; r < 8; ++r) {
    const float di = dinv[m0 + r + half * 8];
    w2s[r] = di * di;
  }

  // store D tiles: H raw, agg = dinv^2 * H
#pragma unroll
  for (int t = 0; t < 8; ++t) {
#pragma unroll
    for (int r = 0; r < 8; ++r) {
      const size_t idx = (size_t)(m0 + r + half * 8) * HID_DIM + t * 16 + l15;
      const float v = c[t][r];
      H[idx] = v;
      agg[idx] = w2s[r] * v;
    }
  }
}

// ---------------- aggregation layer 1 --------------------------------------
// one wave per edge: gather 128 floats of H[src], scatter-add into agg[dst]
__global__ void __launch_bounds__(256)
k_edge_agg(const int* __restrict__ src, const int* __restrict__ dst,
           const float* __restrict__ dinv, const float* __restrict__ H,
           float* __restrict__ agg, int E) {
  const int wid = (blockIdx.x * blockDim.x + threadIdx.x) >> 5;
  const int lane = threadIdx.x & 31;
  if (wid >= E) return;
  const int s = src[wid];
  const int d = dst[wid];
  const float w = dinv[s] * dinv[d];
  const float4 hv = ((const float4*)(H + (size_t)s * HID_DIM))[lane];
  float* ap = agg + (size_t)d * HID_DIM + lane * 4;
  atomicAdd(ap + 0, w * hv.x);
  atomicAdd(ap + 1, w * hv.y);
  atomicAdd(ap + 2, w * hv.z);
  atomicAdd(ap + 3, w * hv.w);
}

// ---------------- layer 2: z = relu(agg + b1) @ W2 (128-dot per node) ------
// one wave per node; bias+ReLU fused; also writes self-loop term of output
__global__ void __launch_bounds__(256)
k_node_dot(const float* __restrict__ agg, const float* __restrict__ b1,
           const float* __restrict__ w2, const float* __restrict__ dinv,
           float* __restrict__ z, float* __restrict__ out, int N) {
  const int wid = (blockIdx.x * blockDim.x + threadIdx.x) >> 5;
  const int lane = threadIdx.x & 31;
  if (wid >= N) return;
  float4 hv = ((const float4*)(agg + (size_t)wid * HID_DIM))[lane];
  const float4 bv = ((const float4*)b1)[lane];
  const float4 wv = ((const float4*)w2)[lane];
  hv.x = fmaxf(hv.x + bv.x, 0.f);
  hv.y = fmaxf(hv.y + bv.y, 0.f);
  hv.z = fmaxf(hv.z + bv.z, 0.f);
  hv.w = fmaxf(hv.w + bv.w, 0.f);
  float p = hv.x * wv.x + hv.y * wv.y + hv.z * wv.z + hv.w * wv.w;
#pragma unroll
  for (int off = 16; off > 0; off >>= 1) p += __shfl_down(p, off, 32);
  if (lane == 0) {
    z[wid] = p;
    const float di = dinv[wid];
    out[wid] = di * di * p;  // self-loop contribution; also initializes d_out
  }
}

// scalar scatter-add per edge
__global__ void k_edge_scalar(const int* __restrict__ src, const int* __restrict__ dst,
                              const float* __restrict__ dinv, const float* __restrict__ z,
                              float* __restrict__ out, int E) {
  int e = blockIdx.x * blockDim.x + threadIdx.x;
  if (e >= E) return;
  const int s = src[e];
  const int d = dst[e];
  atomicAdd(out + d, dinv[s] * dinv[d] * z[s]);
}

// out = mask ? 0 : out + b2
__global__ void k_finalize(float* __restrict__ out, const float* __restrict__ b2,
                           const int* __restrict__ mask, int N) {
  int i = blockIdx.x * blockDim.x + threadIdx.x;
  if (i >= N) return;
  out[i] = mask[i] ? 0.f : (out[i] + b2[0]);
}

// ---------------------------------------------------------------------------
extern "C" void kernel_launch(void* const* d_in, const int* in_sizes, int n_in,
                              void* d_out, int out_size, void* d_ws, size_t ws_size,
                              hipStream_t stream) {
  const float* X    = (const float*)d_in[0];
  const int*   eidx = (const int*)d_in[1];      // [2, E] row-major: row0=src, row1=dst
  const int*   mask = (const int*)d_in[2];
  const float* W1   = (const float*)d_in[3];
  const float* b1   = (const float*)d_in[4];
  const float* W2   = (const float*)d_in[5];
  const float* b2   = (const float*)d_in[6];
  float* out = (float*)d_out;

  const int N = in_sizes[0] / IN_DIM;
  const int E = in_sizes[1] / 2;
  const int* src = eidx;
  const int* dst = eidx + E;

  // workspace layout (float granularity, 256B aligned chunks)
  const size_t Npad = ((size_t)N + 63) & ~(size_t)63;
  float* dinv = (float*)d_ws;             // N
  float* H    = dinv + Npad;              // N*128
  float* agg  = H + Npad * HID_DIM;       // N*128
  float* z    = agg + Npad * HID_DIM;     // N

  const int T = 256;

  // 1) degree (with self-loops folded in) -> dinv
  k_deg_init<<<(N + T - 1) / T, T, 0, stream>>>(dinv, N);
  k_deg_accum<<<(E + T - 1) / T, T, 0, stream>>>(dst, dinv, E);
  k_rsqrt_inplace<<<(N + T - 1) / T, T, 0, stream>>>(dinv, N);

  // 2) H = X @ W1 (WMMA f32 16x16x4); epilogue seeds agg with self-loop term
  {
    const int rowTiles = (N + 15) / 16;
    const int blocks = (rowTiles + 3) / 4;  // 4 waves / block
    k_gemm_x_w1<<<blocks, 128, 0, stream>>>(X, W1, dinv, H, agg, N);
  }

  // 3) aggregate layer 1 (atomics into L2-resident agg)
  {
    const int waves_per_block = T / 32;
    k_edge_agg<<<(E + waves_per_block - 1) / waves_per_block, T, 0, stream>>>(
        src, dst, dinv, H, agg, E);
  }

  // 4) layer 2: fused bias+relu+dot per node, scalar edge scatter, finalize
  {
    const int waves_per_block = T / 32;
    k_node_dot<<<(N + waves_per_block - 1) / waves_per_block, T, 0, stream>>>(
        agg, b1, W2, dinv, z, out, N);
    k_edge_scalar<<<(E + T - 1) / T, T, 0, stream>>>(src, dst, dinv, z, out, E);
    k_finalize<<<(N + T - 1) / T, T, 0, stream>>>(out, b2, mask, N);
  }
}